// FFT_22574348108481
// MI455X (gfx1250) — compile-verified
//
#include <hip/hip_runtime.h>
#include <math.h>

// ---------------------------------------------------------------------------
// GFNet spectral filter for MI455X (gfx1250, wave32, WMMA f32 16x16x4).
// out = irfft2( rfft2(x, ortho) * W, ortho ) on 14x14 spatial grids,
// x: (B=256, 196, C=768) fp32, W: (14, 8, C, 2) fp32.
// One wave per (batch, 16-channel tile). X tile DMA'd into LDS by the Tensor
// Data Mover (tensor_load_to_lds / s_wait_tensorcnt) while the wave builds
// constant DFT tables. All DFT stages run on the f32 WMMA pipe; inter-stage
// buffers use a fragment-native paired-row layout so every B-fragment is one
// ds_load_b64 and D-stores pair into ds_store_b64.
// ---------------------------------------------------------------------------

typedef __attribute__((ext_vector_type(2))) float v2f;
typedef __attribute__((ext_vector_type(8))) float v8f;
typedef __attribute__((ext_vector_type(4))) unsigned int u32x4;
typedef __attribute__((ext_vector_type(8))) int i32x8;
typedef __attribute__((ext_vector_type(4))) int i32x4;

__device__ __forceinline__ v8f wmma4(v2f a, v2f b, v8f c) {
  return __builtin_amdgcn_wmma_f32_16x16x4_f32(false, a, false, b, (short)0, c,
                                               false, false);
}

// A-fragment (16x4 f32): lanes 0-15 hold {K0,K1}, lanes 16-31 hold {K2,K3},
// M = lane % 16; the two elements are adjacent -> one ds_load_b64.
__device__ __forceinline__ v2f load_afrag(const float* T, int ldk, int lane,
                                          int k0) {
  const int m  = lane & 15;
  const int kh = (lane >> 4) << 1;
  return *(const v2f*)&T[m * ldk + k0 + kh];
}

#define HS 14
#define WS 14
#define WH 8     // rfft width
#define CT 16    // channel tile
#define NCT 48   // 768 / 16

// Paired-row index for the stacked [Re(0..13); Im(0..13)] buffers:
// element (r, n) lives at [r>>2][r&1][n][(r>>1)&1] so that a B-fragment
// lane (kb = lane>>4) reads rows {4s+kb, 4s+2+kb} as one contiguous v2f.
__device__ __forceinline__ int pidx(int r, int n) {
  return (((r >> 2) << 6) | ((r & 1) << 5) | (n << 1) | ((r >> 1) & 1));
}

__global__ __launch_bounds__(32) void gfnet_filter_kernel(
    const float* __restrict__ x, const float* __restrict__ cw,
    float* __restrict__ out, int C) {
  // ---- LDS ----------------------------------------------------------------
  __shared__ __align__(16) float A1[16 * 16];  // rfft-W : rows 0-7 cos, 8-15 -sin
  __shared__ __align__(16) float A2[32 * 32];  // fft-H  : [[C,S],[-S,C]] (28x28)
  __shared__ __align__(16) float A4[32 * 32];  // ifft-H : [[C,-S],[S,C]]
  __shared__ __align__(16) float A5[16 * 16];  // irfft-W: amp*cos | -amp*sin
  __shared__ __align__(16) float Xs[HS * WS * CT];  // [h][w][c] (TDM target)
  __shared__ __align__(16) float Ys[7 * 512];  // paired: [k>>2][k&1][wp][n][slot]
  __shared__ __align__(16) float Zs[WH * 512]; // per w': paired rows [Zr;Zi]

  const int lane = threadIdx.x;
  const int b    = blockIdx.x / NCT;
  const int c0   = (blockIdx.x % NCT) * CT;
  const float SN  = 0.26726124191242438f;  // 1/sqrt(14), ortho per 1D axis
  const float W14 = 0.44879895051282760f;  // 2*pi/14

  // ---- kick off TDM: DMA the 196x16 fp32 tile (196 rows x 64B) into Xs ---
  {
    const unsigned long long ga =
        (unsigned long long)(size_t)(x + ((size_t)b * 196) * C + c0);
    const unsigned lds_base = (unsigned)(size_t)(&Xs[0]);
    u32x4 g0;
    g0[0] = 1u;                                 // count=1, user mode
    g0[1] = lds_base;                           // D#.lds_addr
    g0[2] = (unsigned)(ga & 0xFFFFFFFFu);       // D#.global_addr[31:0]
    g0[3] = (unsigned)((ga >> 32) & 0x01FFFFFFu) | (2u << 30);  // [56:32]|type=2
    i32x8 g1;
    g1[0] = (int)(2u << 16);     // data_size=4B, workgroup_mask=0
    g1[1] = (int)(768u << 16);   // tensor_dim0 = 768 (bits 63:48)
    g1[2] = (int)(196u << 16);   // tensor_dim1 = 196 (bits 95:80)
    g1[3] = (int)(16u << 16);    // tile_dim0   = 16  (bits 127:112)
    g1[4] = 196;                 // tile_dim1   = 196 (bits 143:128)
    g1[5] = 768;                 // tensor_dim0_stride = 768 (bits 191:160)
    g1[6] = 0;
    g1[7] = 0;
    i32x4 gz4 = {0, 0, 0, 0};              // groups 2/3 unused (2D tile)
    i32x8 gz8 = {0, 0, 0, 0, 0, 0, 0, 0};  // extra group (6-arg toolchain form)
    __builtin_amdgcn_tensor_load_to_lds(g0, g1, gz4, gz4, gz8, 0);
  }

  // ---- build constant DFT tables (overlaps the TDM transfer) -------------
  for (int i = lane; i < 256; i += 32) {
    const int m = i >> 4, k = i & 15;
    float v1 = 0.0f;
    if (k < WS)
      v1 = (m < WH) ? SN * cosf(W14 * (float)(m * k))
                    : -SN * sinf(W14 * (float)((m - WH) * k));
    A1[i] = v1;
    float v5;
    if (k < WH) {
      const float amp = (k == 0 || k == 7) ? 1.0f : 2.0f;
      v5 = SN * amp * cosf(W14 * (float)(m * k));
    } else {
      const int q = k - WH;
      const float amp = (q == 0 || q == 7) ? 1.0f : 2.0f;
      v5 = -SN * amp * sinf(W14 * (float)(m * q));
    }
    A5[i] = v5;
  }
  for (int i = lane; i < 1024; i += 32) {
    const int m = i >> 5, k = i & 31;
    float v2 = 0.0f, v4 = 0.0f;
    if (m < 28 && k < 28) {
      const int mr = (m < HS) ? m : m - HS;
      const int kr = (k < HS) ? k : k - HS;
      const float cc = SN * cosf(W14 * (float)(mr * kr));
      const float ss = SN * sinf(W14 * (float)(mr * kr));
      const bool mi = (m >= HS), ki = (k >= HS);
      if (!mi && !ki)      { v2 =  cc; v4 =  cc; }
      else if (!mi && ki)  { v2 =  ss; v4 = -ss; }
      else if (mi && !ki)  { v2 = -ss; v4 =  ss; }
      else                 { v2 =  cc; v4 =  cc; }
    }
    A2[i] = v2;
    A4[i] = v4;
  }
  __syncthreads();

  const int n  = lane & 15;         // N column of B/C/D fragments
  const int kb = lane >> 4;         // B-fragment K selector (0 or 1)
  const int hi = lane >> 4;         // D-fragment half
  const int ro = hi * 8;            // D-fragment row offset

  // ---- hoist all constant A-fragments into VGPRs -------------------------
  v2f a1f[4], a5f[4], a2f[2][7], a4f[2][7];
#pragma unroll
  for (int s = 0; s < 4; ++s) {
    a1f[s] = load_afrag(A1, 16, lane, s * 4);
    a5f[s] = load_afrag(A5, 16, lane, s * 4);
  }
#pragma unroll
  for (int s = 0; s < 7; ++s) {
    a2f[0][s] = load_afrag(A2, 32, lane, s * 4);
    a2f[1][s] = load_afrag(A2 + 16 * 32, 32, lane, s * 4);
    a4f[0][s] = load_afrag(A4, 32, lane, s * 4);
    a4f[1][s] = load_afrag(A4 + 16 * 32, 32, lane, s * 4);
  }

  // ---- wait for the TDM tile, then run the GEMM pipeline -----------------
  __builtin_amdgcn_s_wait_tensorcnt(0);
  __syncthreads();

  // ---- stage 1: rfft over W, per h ---------------------------------------
  for (int h = 0; h < HS; ++h) {
    v8f acc = {};
    const float* Bb = &Xs[h * (WS * CT)];
#pragma unroll
    for (int s = 0; s < 4; ++s) {
      v2f bf;
      bf.x = (s * 4 + 1 < WS) ? Bb[(s * 4 + kb) * CT + n] : 0.0f;
      bf.y = (s * 4 + 3 < WS) ? Bb[(s * 4 + 2 + kb) * CT + n] : 0.0f;
      acc = wmma4(a1f[s], bf, acc);
    }
    // D rows p = vi + 8*hi: wp = vi, k = h + 14*hi in the stacked [Yr;Yi].
    const int kk = h + HS * hi;
    const int sb = ((kk >> 2) << 9) | ((kk & 1) << 8) | (n << 1) | ((kk >> 1) & 1);
#pragma unroll
    for (int vi = 0; vi < 8; ++vi) Ys[sb + (vi << 5)] = acc[vi];
  }
  __syncthreads();

  // ---- stage 2: forward fft over H, per w' (28x28 real form) -------------
  for (int wp = 0; wp < WH; ++wp) {
    v2f bfr[7];
#pragma unroll
    for (int s = 0; s < 7; ++s)  // one ds_load_b64 per K-step
      bfr[s] = *(const v2f*)&Ys[(s << 9) | (kb << 8) | (wp << 5) | (n << 1)];
    v8f acc0 = {}, acc1 = {};
#pragma unroll
    for (int s = 0; s < 7; ++s) {
      acc0 = wmma4(a2f[0][s], bfr[s], acc0);
      acc1 = wmma4(a2f[1][s], bfr[s], acc1);
    }
    float* Zw = &Zs[wp << 9];
#pragma unroll
    for (int vp = 0; vp < 4; ++vp) {         // paired ds_store_b64
      const int vi = (vp & 1) | ((vp >> 1) << 2);  // 0,1,4,5
      const int r0 = ro + vi, r1 = ro + vi + 16;
      v2f s0; s0.x = acc0[vi]; s0.y = acc0[vi + 2];
      v2f s1; s1.x = acc1[vi]; s1.y = acc1[vi + 2];
      *(v2f*)&Zw[((r0 >> 2) << 6) | ((r0 & 1) << 5) | (n << 1)] = s0;
      *(v2f*)&Zw[((r1 >> 2) << 6) | ((r1 & 1) << 5) | (n << 1)] = s1;
    }
  }
  __syncthreads();

  // ---- stage 3: pointwise complex weight (stays L2-resident) -------------
  for (int i = lane; i < HS * WH * CT; i += 32) {
    const int c  = i & 15;
    const int wp = (i >> 4) & 7;
    const int hp = i >> 7;
    float* Zw = &Zs[wp << 9];
    const int ir = pidx(hp, c), ii = pidx(HS + hp, c);
    const float zr = Zw[ir];
    const float zi = Zw[ii];
    const v2f wv = *(const v2f*)(cw + (((size_t)(hp * WH + wp)) * C + c0 + c) * 2);
    Zw[ir] = zr * wv.x - zi * wv.y;
    Zw[ii] = zr * wv.y + zi * wv.x;
  }
  __syncthreads();

  // ---- stage 4: inverse fft over H, per w', in-place ---------------------
  for (int wp = 0; wp < WH; ++wp) {
    float* Zw = &Zs[wp << 9];
    v2f bfr[7];
#pragma unroll
    for (int s = 0; s < 7; ++s)  // load all B fragments before any store
      bfr[s] = *(const v2f*)&Zw[(s << 6) | (kb << 5) | (n << 1)];
    v8f acc0 = {}, acc1 = {};
#pragma unroll
    for (int s = 0; s < 7; ++s) {
      acc0 = wmma4(a4f[0][s], bfr[s], acc0);
      acc1 = wmma4(a4f[1][s], bfr[s], acc1);
    }
#pragma unroll
    for (int vp = 0; vp < 4; ++vp) {
      const int vi = (vp & 1) | ((vp >> 1) << 2);  // 0,1,4,5
      const int r0 = ro + vi, r1 = ro + vi + 16;
      v2f s0; s0.x = acc0[vi]; s0.y = acc0[vi + 2];
      v2f s1; s1.x = acc1[vi]; s1.y = acc1[vi + 2];
      *(v2f*)&Zw[((r0 >> 2) << 6) | ((r0 & 1) << 5) | (n << 1)] = s0;
      *(v2f*)&Zw[((r1 >> 2) << 6) | ((r1 & 1) << 5) | (n << 1)] = s1;
    }
  }
  __syncthreads();

  // ---- stage 5: irfft over W, per h, store to global ---------------------
  // B rows k: k<8 -> Ur[w'=k] (r=h), k>=8 -> Ui[w'=k-8] (r=14+h).
  for (int h = 0; h < HS; ++h) {
    v8f acc = {};
#pragma unroll
    for (int s = 0; s < 4; ++s) {
      v2f bf;
      bf.x = (s * 4 + 1 < WH)
                 ? Zs[((s * 4 + kb) << 9) + pidx(h, n)]
                 : Zs[((s * 4 + kb - WH) << 9) + pidx(HS + h, n)];
      bf.y = (s * 4 + 3 < WH)
                 ? Zs[((s * 4 + 2 + kb) << 9) + pidx(h, n)]
                 : Zs[((s * 4 + 2 + kb - WH) << 9) + pidx(HS + h, n)];
      acc = wmma4(a5f[s], bf, acc);
    }
    float* ob = out + ((size_t)b * 196 + h * WS) * C + c0 + n;
#pragma unroll
    for (int vi = 0; vi < 8; ++vi) {
      const int w = vi + ro;
      if (w < WS) ob[(size_t)w * C] = acc[vi];  // only rows 14,15 masked out
    }
  }
}

extern "C" void kernel_launch(void* const* d_in, const int* in_sizes, int n_in,
                              void* d_out, int out_size, void* d_ws,
                              size_t ws_size, hipStream_t stream) {
  const float* x  = (const float*)d_in[0];
  const float* cw = (const float*)d_in[1];
  float* out = (float*)d_out;
  const int C = 768;
  const int B = in_sizes[0] / (196 * C);
  dim3 grid(B * NCT);
  gfnet_filter_kernel<<<grid, 32, 0, stream>>>(x, cw, out, C);
}